// DisenIDP_87935160418633
// MI455X (gfx1250) — compile-verified
//
#include <hip/hip_runtime.h>

#define D 64

typedef __attribute__((ext_vector_type(2))) float v2f;
typedef __attribute__((ext_vector_type(8))) float v8f;

// ---------------------------------------------------------------------------
// Self-gating: u = emb * sigmoid(emb @ W + b), also acc = u (layer-0 of the
// propagation accumulator). One wave computes a 16x64 tile via 4x f32 WMMA
// accumulators, K=64 in 16 steps of V_WMMA_F32_16X16X4_F32 (fp32-exact).
// ---------------------------------------------------------------------------
__global__ __launch_bounds__(128)
void disen_gate_wmma(const float* __restrict__ emb,
                     const float* __restrict__ W,
                     const float* __restrict__ bias,
                     float* __restrict__ u,
                     float* __restrict__ acc,
                     int nnode) {
  const int wave  = threadIdx.x >> 5;
  const int lane  = threadIdx.x & 31;
  const int mbase = (blockIdx.x * 4 + wave) * 16;   // 16 nodes per wave
  if (mbase >= nnode) return;                        // wave-uniform guard
  const int mrow  = lane & 15;
  const int khalf = lane >> 4;                       // 0 or 1
  const int ncol  = lane & 15;

  v8f c0 = {}, c1 = {}, c2 = {}, c3 = {};
  for (int k0 = 0; k0 < D; k0 += 4) {
    const int ka = k0 + khalf * 2;
    // A fragment: 16x4 f32; lanes 0-15 hold K={k0,k0+1}, lanes 16-31 K={k0+2,k0+3}
    const float* arow = emb + (long)(mbase + mrow) * D + ka;
    v2f a; a.x = arow[0]; a.y = arow[1];
    // B fragments (4x16 f32) for the four 16-column tiles; B[k][n] = W[k*64+n]
    const float* w0 = W + (long)ka * D;
    const float* w1 = W + (long)(ka + 1) * D;
    v2f b0, b1, b2, b3;
    b0.x = w0[ncol];      b0.y = w1[ncol];
    b1.x = w0[ncol + 16]; b1.y = w1[ncol + 16];
    b2.x = w0[ncol + 32]; b2.y = w1[ncol + 32];
    b3.x = w0[ncol + 48]; b3.y = w1[ncol + 48];
    c0 = __builtin_amdgcn_wmma_f32_16x16x4_f32(false, a, false, b0, (short)0, c0, false, false);
    c1 = __builtin_amdgcn_wmma_f32_16x16x4_f32(false, a, false, b1, (short)0, c1, false, false);
    c2 = __builtin_amdgcn_wmma_f32_16x16x4_f32(false, a, false, b2, (short)0, c2, false, false);
    c3 = __builtin_amdgcn_wmma_f32_16x16x4_f32(false, a, false, b3, (short)0, c3, false, false);
  }

  v8f cs[4] = {c0, c1, c2, c3};
  #pragma unroll
  for (int t = 0; t < 4; ++t) {
    #pragma unroll
    for (int r = 0; r < 8; ++r) {
      const int node = mbase + r + 8 * khalf;        // C/D layout: M = r + 8*(lane>=16)
      const int col  = t * 16 + ncol;
      const float g  = cs[t][r] + bias[col];
      const float s  = 1.0f / (1.0f + expf(-g));
      const float o  = emb[(long)node * D + col] * s;
      u[(long)node * D + col]   = o;
      acc[(long)node * D + col] = o;
    }
  }
}

// ---------------------------------------------------------------------------
// Zero-fill (float4 stores)
// ---------------------------------------------------------------------------
__global__ __launch_bounds__(256)
void disen_fill4(float4* __restrict__ p, long n4) {
  long i = (long)blockIdx.x * blockDim.x + threadIdx.x;
  if (i < n4) p[i] = make_float4(0.f, 0.f, 0.f, 0.f);
}

// ---------------------------------------------------------------------------
// Edge-parallel SpMM: y[row] += val * x[col].  16 threads per edge, each
// thread gathers a float4 (global_load_b128, coalesced across the row) and
// scatters with 4 fp32 L2 atomics (global_atomic_add_f32).
// ---------------------------------------------------------------------------
__global__ __launch_bounds__(256)
void disen_spmm(const int* __restrict__ rows, const int* __restrict__ cols,
                const float* __restrict__ vals, const float4* __restrict__ x4,
                float* __restrict__ y, int nedge) {
  long t = (long)blockIdx.x * blockDim.x + threadIdx.x;
  int e = (int)(t >> 4);
  int g = (int)(t & 15);
  if (e >= nedge) return;
  const int r = rows[e];
  const int c = cols[e];
  const float v = vals[e];
  const float4 xv = x4[(long)c * (D / 4) + g];
  float* yp = y + (long)r * D + g * 4;
  unsafeAtomicAdd(yp + 0, v * xv.x);
  unsafeAtomicAdd(yp + 1, v * xv.y);
  unsafeAtomicAdd(yp + 2, v * xv.z);
  unsafeAtomicAdd(yp + 3, v * xv.w);
}

// ---------------------------------------------------------------------------
// acc += y / max(||y||_2, eps), one wave per node (float2 per lane, shfl
// reduce over 32 lanes; reduction order doesn't affect the row sum).
// ---------------------------------------------------------------------------
__global__ __launch_bounds__(256)
void disen_normacc(const float* __restrict__ y, float* __restrict__ acc, int nnode) {
  const int node = blockIdx.x * 8 + (threadIdx.x >> 5);
  const int lane = threadIdx.x & 31;
  if (node >= nnode) return;
  const float2 a = ((const float2*)(y + (long)node * D))[lane];
  float ss = a.x * a.x + a.y * a.y;
  #pragma unroll
  for (int m = 16; m; m >>= 1) ss += __shfl_xor(ss, m, 32);
  const float rinv = 1.0f / fmaxf(sqrtf(ss), 1e-12f);
  float2* ap = (float2*)(acc + (long)node * D);
  float2 o = ap[lane];
  o.x += a.x * rinv;
  o.y += a.y * rinv;
  ap[lane] = o;
}

// ---------------------------------------------------------------------------
// v[k] = sum_j att_m[k][j] * att[j]   (collapses att*(e@att_m) to e.v)
// ---------------------------------------------------------------------------
__global__ void disen_attvec(const float* __restrict__ att,
                             const float* __restrict__ attm,
                             float* __restrict__ v) {
  const int k = threadIdx.x;
  float s = 0.0f;
  #pragma unroll 8
  for (int j = 0; j < D; ++j) s += attm[(long)k * D + j] * att[j];
  v[k] = s;
}

// ---------------------------------------------------------------------------
// Per-node attention softmax + mix.  One wave per node, float2 per lane.
// ---------------------------------------------------------------------------
__global__ __launch_bounds__(256)
void disen_mix(const float* __restrict__ e2, const float* __restrict__ e3,
               const float* __restrict__ v, float* __restrict__ mixed,
               float* __restrict__ score, int nnode) {
  const int node = blockIdx.x * 8 + (threadIdx.x >> 5);
  const int lane = threadIdx.x & 31;
  if (node >= nnode) return;
  const float2 vv = ((const float2*)v)[lane];
  const float2 a  = ((const float2*)(e2 + (long)node * D))[lane];
  const float2 b  = ((const float2*)(e3 + (long)node * D))[lane];
  float w2 = a.x * vv.x + a.y * vv.y;
  float w3 = b.x * vv.x + b.y * vv.y;
  #pragma unroll
  for (int m = 16; m; m >>= 1) {
    w2 += __shfl_xor(w2, m, 32);
    w3 += __shfl_xor(w3, m, 32);
  }
  const float mx  = fmaxf(w2, w3);
  const float ex2 = expf(w2 - mx);
  const float ex3 = expf(w3 - mx);
  const float inv = 1.0f / (ex2 + ex3);
  const float s2 = ex2 * inv, s3 = ex3 * inv;
  float2 o;
  o.x = s2 * a.x + s3 * b.x;
  o.y = s2 * a.y + s3 * b.y;
  ((float2*)(mixed + (long)node * D))[lane] = o;
  if (lane == 0) {
    score[(long)node * 2]     = s2;
    score[(long)node * 2 + 1] = s3;
  }
}

extern "C" void kernel_launch(void* const* d_in, const int* in_sizes, int n_in,
                              void* d_out, int out_size, void* d_ws, size_t ws_size,
                              hipStream_t stream) {
  const float* emb  = (const float*)d_in[0];
  const float* W0   = (const float*)d_in[1];
  const float* b0   = (const float*)d_in[2];
  const float* W1   = (const float*)d_in[3];
  const float* b1   = (const float*)d_in[4];
  const float* att  = (const float*)d_in[5];
  const float* attm = (const float*)d_in[6];
  const int*   ir   = (const int*)d_in[7];
  const int*   ic   = (const int*)d_in[8];
  const float* iv   = (const float*)d_in[9];
  const int*   ur   = (const int*)d_in[10];
  const int*   uc   = (const int*)d_in[11];
  const float* uv   = (const float*)d_in[12];

  const int nnode = in_sizes[0] / D;      // 200000
  const int nedge = in_sizes[7];          // 2000000
  const long total = (long)nnode * D;

  float* ws   = (float*)d_ws;
  float* acc2 = ws;                        // e2 accumulator
  float* acc3 = ws + total;                // e3 accumulator
  float* bufA = ws + 2 * total;            // ping
  float* bufB = ws + 3 * total;            // pong
  float* vvec = ws + 4 * total;            // 64-float attention vector

  float* mixed = (float*)d_out;
  float* score = (float*)d_out + total;

  const int gateBlocks = (nnode + 63) / 64;               // 4 waves x 16 rows
  const int fillBlocks = (int)((total / 4 + 255) / 256);
  const int spmmBlocks = (int)(((long)nedge * 16 + 255) / 256);
  const int waveBlocks = (nnode + 7) / 8;                 // 8 waves/block

  struct Ch { const float* W; const float* b; const int* r; const int* c;
              const float* v; float* acc; };
  const Ch chans[2] = { {W0, b0, ir, ic, iv, acc2},
                        {W1, b1, ur, uc, uv, acc3} };

  for (int ci = 0; ci < 2; ++ci) {
    const Ch ch = chans[ci];
    disen_gate_wmma<<<gateBlocks, 128, 0, stream>>>(emb, ch.W, ch.b, bufA, ch.acc, nnode);
    float* u = bufA;
    float* y = bufB;
    for (int l = 0; l < 3; ++l) {
      disen_fill4<<<fillBlocks, 256, 0, stream>>>((float4*)y, total / 4);
      disen_spmm<<<spmmBlocks, 256, 0, stream>>>(ch.r, ch.c, ch.v, (const float4*)u, y, nedge);
      disen_normacc<<<waveBlocks, 256, 0, stream>>>(y, ch.acc, nnode);
      float* t = u; u = y; y = t;
    }
  }
  disen_attvec<<<1, D, 0, stream>>>(att, attm, vvec);
  disen_mix<<<waveBlocks, 256, 0, stream>>>(acc2, acc3, vvec, mixed, score, nnode);
}